// VNNBlock_16552803959041
// MI455X (gfx1250) — compile-verified
//
#include <hip/hip_runtime.h>

// MI455X / gfx1250, wave32. fp32 throughout (matches reference numerics).
//
// Algebraic collapse: weights[b,i,o] depends on i only via m=(i&1)*256+o
// (both pe1 and xr are indexed by k % IN), so
//   out[b,o] = SE[b]*G[b,o] + SO[b]*G[b,256+o]
// with SE/SO the even/odd-index sums of x[b,:]. The only real matmuls are the
// pos_enc @ W projections -> V_WMMA_F32_16X16X4_F32.

typedef __attribute__((ext_vector_type(2))) float v2f;
typedef __attribute__((ext_vector_type(8))) float v8f;

#define D_MODEL 16
#define HIDDEN  64
#define B_SZ    32
#define IN_SZ   512
#define OUT_SZ  256

// div_j = 10000^(-j/8), j = 0..7  (compile-time; no expf on device)
__device__ __constant__ const float PE_DIV[8] = {
    1.0f,
    0.31622776601683794f,   // 10^-0.5
    0.1f,
    0.031622776601683791f,  // 10^-1.5
    0.01f,
    0.0031622776601683794f, // 10^-2.5
    0.001f,
    0.00031622776601683794f // 10^-3.5
};

// A-fragment pair for even d = 2j: (sin(idx*div_j), cos(idx*div_j))
__device__ __forceinline__ v2f pe_pair(int idx, float divj) {
    float s, c;
    sincosf((float)idx * divj, &s, &c);
    v2f r; r.x = s; r.y = c;
    return r;
}

// ---------------------------------------------------------------------------
// Kernel 1: WMMA build of PP[512][64] (pe1@w1[:16] + pe2@w1[16:32] + b1) and
//           BQ[256][64] (pos_enc(o)@bw1[:16] + bb1).
// One wave per block. Blocks 0..127 -> PP tiles (32 m-tiles x 4 n-tiles),
// blocks 128..191 -> BQ tiles (16 o-tiles x 4 n-tiles).
//
// V_WMMA_F32_16X16X4_F32 operand layout (ISA 7.12.2, 32-bit matrices):
//   A 16x4 : lane L holds (M = L&15, K = (L>>4)*2 + {vgpr0,vgpr1})
//   B 4x16 : lane L holds (N = L&15, K = (L>>4)*2 + {vgpr0,vgpr1})
//   C/D    : vgpr r -> (M = 8*(L>>4) + r, N = L&15)
// ---------------------------------------------------------------------------
__global__ __launch_bounds__(32) void vnn_pe_wmma(
    const float* __restrict__ w1, const float* __restrict__ b1,
    const float* __restrict__ bw1, const float* __restrict__ bb1,
    float* __restrict__ PP, float* __restrict__ BQ)
{
    const int l    = threadIdx.x;
    const int half = l >> 4;   // 0 or 1: selects K pair {0,1} vs {2,3}
    const int lm   = l & 15;

    v8f c = {0.f, 0.f, 0.f, 0.f, 0.f, 0.f, 0.f, 0.f};
    const int blk = blockIdx.x;

    if (blk < 128) {
        // ---- PP tile ----
        const int mtile = blk >> 2;
        const int ntile = blk & 3;
        const int mrow  = mtile * 16 + lm;   // A-matrix row (M) for this lane
        const int n     = ntile * 16 + lm;   // B-matrix col (N) for this lane
        #pragma unroll
        for (int pass = 0; pass < 2; ++pass) {
            const int idx   = pass ? (mrow & (OUT_SZ - 1)) : mrow; // pe1: m, pe2: m%256
            const int rbase = pass * D_MODEL;                      // w1 rows 0..15 / 16..31
            #pragma unroll
            for (int kk = 0; kk < 4; ++kk) {                       // K=16 in 4 steps of 4
                const int   k0   = kk * 4 + half * 2;              // even
                const float divj = half ? PE_DIV[kk * 2 + 1] : PE_DIV[kk * 2];
                v2f a = pe_pair(idx, divj);
                v2f bm;
                bm.x = w1[(rbase + k0)     * HIDDEN + n];
                bm.y = w1[(rbase + k0 + 1) * HIDDEN + n];
                c = __builtin_amdgcn_wmma_f32_16x16x4_f32(
                        false, a, false, bm, (short)0, c, false, false);
            }
        }
        const float bn = b1[n];
        #pragma unroll
        for (int r = 0; r < 8; ++r) {
            const int M = mtile * 16 + half * 8 + r;
            PP[M * HIDDEN + n] = c[r] + bn;
        }
    } else {
        // ---- BQ tile ----
        const int blk2  = blk - 128;
        const int otile = blk2 >> 2;
        const int ntile = blk2 & 3;
        const int orow  = otile * 16 + lm;
        const int n     = ntile * 16 + lm;
        #pragma unroll
        for (int kk = 0; kk < 4; ++kk) {
            const int   k0   = kk * 4 + half * 2;
            const float divj = half ? PE_DIV[kk * 2 + 1] : PE_DIV[kk * 2];
            v2f a = pe_pair(orow, divj);
            v2f bm;
            bm.x = bw1[k0       * HIDDEN + n];
            bm.y = bw1[(k0 + 1) * HIDDEN + n];
            c = __builtin_amdgcn_wmma_f32_16x16x4_f32(
                    false, a, false, bm, (short)0, c, false, false);
        }
        const float bn = bb1[n];
        #pragma unroll
        for (int r = 0; r < 8; ++r) {
            const int O = otile * 16 + half * 8 + r;
            BQ[O * HIDDEN + n] = c[r] + bn;
        }
    }
}

// ---------------------------------------------------------------------------
// Kernel 2: per (b,o): out0 = SE*G(o) + SO*G(256+o), then bias MLP, final out.
// One block per batch (256 threads = 256 outputs).
// ---------------------------------------------------------------------------
__global__ __launch_bounds__(256) void vnn_out(
    const float* __restrict__ x,  const float* __restrict__ w1,
    const float* __restrict__ w2, const float* __restrict__ b2,
    const float* __restrict__ bw1, const float* __restrict__ bw2,
    const float* __restrict__ bb2,
    const float* __restrict__ PP, const float* __restrict__ BQ,
    float* __restrict__ out)
{
    const int b = blockIdx.x;
    const int o = threadIdx.x;

    __shared__ float sE[OUT_SZ];
    __shared__ float sO[OUT_SZ];

    // even/odd index sums of x[b,:]
    sE[o] = x[b * IN_SZ + 2 * o];
    sO[o] = x[b * IN_SZ + 2 * o + 1];
    __syncthreads();
    #pragma unroll
    for (int s = 128; s > 0; s >>= 1) {
        if (o < s) { sE[o] += sE[o + s]; sO[o] += sO[o + s]; }
        __syncthreads();
    }
    const float SE = sE[0];
    const float SO = sO[0];

    // prefetch the two PP rows this lane walks (global_prefetch_b8)
    __builtin_prefetch(&PP[o * HIDDEN], 0, 1);
    __builtin_prefetch(&PP[(OUT_SZ + o) * HIDDEN], 0, 1);

    const float tE  = x[b * IN_SZ + o];            // x[b, m] with m = o
    const float tO  = x[b * IN_SZ + OUT_SZ + o];   // x[b, m] with m = 256 + o
    const float b2v = b2[0];

    float gE = 0.f, gO = 0.f;
    #pragma unroll
    for (int h = 0; h < HIDDEN; ++h) {
        const float vh = w1[2 * D_MODEL * HIDDEN + h];  // w1[32,:] (uniform -> SGPR)
        const float wh = w2[h];                          // uniform -> SGPR
        gE = fmaf(wh, fmaxf(fmaf(tE, vh, PP[o * HIDDEN + h]), 0.f), gE);
        gO = fmaf(wh, fmaxf(fmaf(tO, vh, PP[(OUT_SZ + o) * HIDDEN + h]), 0.f), gO);
    }
    const float out0 = SE * (gE + b2v) + SO * (gO + b2v);

    // bias_nn: relu(BQ[o,:] + out0*bw1[16,:]) @ bw2 + bb2
    float acc = 0.f;
    #pragma unroll
    for (int h = 0; h < HIDDEN; ++h) {
        const float bvh = bw1[D_MODEL * HIDDEN + h];     // bw1[16,:]
        acc = fmaf(bw2[h], fmaxf(fmaf(out0, bvh, BQ[o * HIDDEN + h]), 0.f), acc);
    }
    out[b * OUT_SZ + o] = out0 + acc + bb2[0];
}

extern "C" void kernel_launch(void* const* d_in, const int* in_sizes, int n_in,
                              void* d_out, int out_size, void* d_ws, size_t ws_size,
                              hipStream_t stream) {
    (void)in_sizes; (void)n_in; (void)out_size; (void)ws_size;
    const float* x   = (const float*)d_in[0];
    const float* w1  = (const float*)d_in[1];
    const float* b1  = (const float*)d_in[2];
    const float* w2  = (const float*)d_in[3];
    const float* b2  = (const float*)d_in[4];
    const float* bw1 = (const float*)d_in[5];
    const float* bb1 = (const float*)d_in[6];
    const float* bw2 = (const float*)d_in[7];
    const float* bb2 = (const float*)d_in[8];
    float* out = (float*)d_out;

    float* PP = (float*)d_ws;              // 512*64 f32 = 128 KB
    float* BQ = PP + IN_SZ * HIDDEN;       // 256*64 f32 =  64 KB

    vnn_pe_wmma<<<192, 32, 0, stream>>>(w1, b1, bw1, bb1, PP, BQ);
    vnn_out<<<B_SZ, OUT_SZ, 0, stream>>>(x, w1, w2, b2, bw1, bw2, bb2, PP, BQ, out);
}